// QuIPLinear_72052371357901
// MI455X (gfx1250) — compile-verified
//
#include <hip/hip_runtime.h>

typedef __attribute__((ext_vector_type(2))) float v2f;
typedef __attribute__((ext_vector_type(8))) float v8f;

#define LDL 36  // k_lora LDS row stride (32 data + 4 pad; 16B-aligned rows)
#define LDG 20  // k_gemm LDS row stride (16 data + 4 pad; 16B-aligned rows)

__device__ __forceinline__ v8f wmma_f32(v2f a, v2f b, v8f c) {
  // D = A(16x4) * B(4x16) + C(16x16), full fp32
  return __builtin_amdgcn_wmma_f32_16x16x4_f32(false, a, false, b, (short)0, c,
                                               false, false);
}

// CDNA5 async global->LDS copy: 16 bytes per lane, tracked by ASYNCcnt.
// VDST = per-lane LDS byte offset (low 32 bits of a generic shared pointer:
// the shared aperture lives in addr[63:32], addr[31:0] is the LDS offset).
__device__ __forceinline__ void async_load_f4(const float* lds_dst,
                                              const float* gsrc) {
  unsigned lds_off = (unsigned)(unsigned long long)lds_dst;
  asm volatile("global_load_async_to_lds_b128 %0, %1, off"
               :
               : "v"(lds_off), "v"(gsrc)
               : "memory");
}

__device__ __forceinline__ void wait_async0() {
#if __has_builtin(__builtin_amdgcn_s_wait_asynccnt)
  __builtin_amdgcn_s_wait_asynccnt(0);
#else
  asm volatile("s_wait_asynccnt 0x0" ::: "memory");
#endif
}

// ---------------------------------------------------------------------------
// Kernel 1: per-row sign flip + fast Walsh-Hadamard (block 128) + smooth +
// blockwise (16) symmetric INT4 fake quantization.
// One 256-thread workgroup per row of x (row length D).
// ---------------------------------------------------------------------------
__global__ __launch_bounds__(256) void k_prep(
    const float* __restrict__ x, const float* __restrict__ S_in,
    const float* __restrict__ smooth, float* __restrict__ x_q,
    float* __restrict__ x_sm, int D) {
  extern __shared__ float xs[];
  const int m = blockIdx.x;
  const int t = threadIdx.x;
  const float* xrow = x + (size_t)m * D;

  for (int i = t; i < D; i += 256) xs[i] = xrow[i] * S_in[i];
  __syncthreads();

  // FWHT over each 128-block (H is symmetric Sylvester: x@H == FWHT(x))
  const int totPairs = D >> 1;
  for (int s = 0; s < 7; ++s) {
    const int h = 1 << s;
    for (int p = t; p < totPairs; p += 256) {
      const int blk = p >> 6;
      const int q = p & 63;
      const int low = ((q >> s) << (s + 1)) + (q & (h - 1));
      const int i0 = blk * 128 + low;
      const float a = xs[i0];
      const float b = xs[i0 + h];
      xs[i0] = a + b;
      xs[i0 + h] = a - b;
    }
    __syncthreads();
  }

  const float norm = 0.08838834764831845f;  // 1/sqrt(128)
  for (int i = t; i < D; i += 256) xs[i] = xs[i] * norm * smooth[i];
  __syncthreads();

  const int nq = D >> 4;
  for (int bq = t; bq < nq; bq += 256) {
    float v[16];
    float amax = 0.f;
#pragma unroll
    for (int j = 0; j < 16; ++j) {
      v[j] = xs[bq * 16 + j];
      amax = fmaxf(amax, fabsf(v[j]));
    }
    amax = fmaxf(amax, 1e-12f);
    const float scale = amax * (1.0f / 7.0f);
    const float inv = 7.0f / amax;
    float qv[16];
#pragma unroll
    for (int j = 0; j < 16; ++j) {
      float q = rintf(v[j] * inv);  // round-half-even == jnp.round
      q = fminf(fmaxf(q, -7.f), 7.f);
      qv[j] = q * scale;
    }
    const size_t base = (size_t)m * D + (size_t)bq * 16;
    float4* q4 = (float4*)(x_q + base);
    float4* s4 = (float4*)(x_sm + base);
#pragma unroll
    for (int j = 0; j < 4; ++j) {
      q4[j] = make_float4(qv[4 * j], qv[4 * j + 1], qv[4 * j + 2], qv[4 * j + 3]);
      s4[j] = make_float4(v[4 * j], v[4 * j + 1], v[4 * j + 2], v[4 * j + 3]);
    }
  }
}

// ---------------------------------------------------------------------------
// Kernel 2: t = x_smooth @ lora_a^T   (M x 32), WMMA f32 16x16x4.
// WG = 256 threads (8 waves); tile = 128 rows x 32 cols.
// ---------------------------------------------------------------------------
__global__ __launch_bounds__(256) void k_lora(
    const float* __restrict__ x_sm, const float* __restrict__ lora_a,
    float* __restrict__ tbuf, int D) {
  __shared__ float Al[128 * LDL];
  __shared__ float Bl[32 * LDL];
  const int t = threadIdx.x;
  const int lane = t & 31;
  const int w = t >> 5;
  const int m0 = blockIdx.x * 128;

  v8f acc0 = {};
  v8f acc1 = {};

  const int nChunks = D / 32;
  const int r = t >> 1;
  const int half = t & 1;
  for (int c = 0; c < nChunks; ++c) {
    const int k0 = c * 32;
    {  // stage A: 128 rows x 32 K
      const float4* src =
          (const float4*)(x_sm + (size_t)(m0 + r) * D + k0 + half * 16);
      float4* dst = (float4*)(Al + r * LDL + half * 16);
#pragma unroll
      for (int j = 0; j < 4; ++j) dst[j] = src[j];
    }
    if (t < 64) {  // stage B: 32 lora_a rows x 32 K
      const float4* src =
          (const float4*)(lora_a + (size_t)r * D + k0 + half * 16);
      float4* dst = (float4*)(Bl + r * LDL + half * 16);
#pragma unroll
      for (int j = 0; j < 4; ++j) dst[j] = src[j];
    }
    __syncthreads();

    const int mrow = w * 16 + (lane & 15);
    const int kb0 = (lane >> 4) << 1;
    const int n = lane & 15;
#pragma unroll
    for (int kk = 0; kk < 32; kk += 4) {
      const int kb = kk + kb0;
      v2f a, b0, b1;
      a.x = Al[mrow * LDL + kb];
      a.y = Al[mrow * LDL + kb + 1];
      b0.x = Bl[n * LDL + kb];
      b0.y = Bl[n * LDL + kb + 1];
      b1.x = Bl[(16 + n) * LDL + kb];
      b1.y = Bl[(16 + n) * LDL + kb + 1];
      acc0 = wmma_f32(a, b0, acc0);
      acc1 = wmma_f32(a, b1, acc1);
    }
    __syncthreads();
  }

  const int n = lane & 15;
  const int mbase = m0 + w * 16 + ((lane >> 4) << 3);
#pragma unroll
  for (int e = 0; e < 8; ++e) {
    const size_t off = (size_t)(mbase + e) * 32;
    tbuf[off + n] = acc0[e];
    tbuf[off + 16 + n] = acc1[e];
  }
}

// ---------------------------------------------------------------------------
// Kernel 3: out = x_q @ w^T  (+ t @ lora_b^T as 2 extra K-chunks) + bias.
// WG = 256 threads (8 waves); output tile 128x128; wave (wm,wn) in 4x2 grid
// owns a 32(M) x 64(N) sub-tile = 2x4 WMMA f32 16x16x4 accumulators.
// K loop: KC=16 chunks, double-buffered LDS filled with CDNA5 async
// global->LDS B128 copies (ASYNCcnt), overlapping HBM with WMMA.
// ---------------------------------------------------------------------------
__global__ __launch_bounds__(256) void k_gemm(
    const float* __restrict__ x_q, const float* __restrict__ tbuf,
    const float* __restrict__ wq, const float* __restrict__ lora_b,
    const float* __restrict__ bias, float* __restrict__ out, int O, int D,
    int R) {
  __shared__ float Al[2][128 * LDG];
  __shared__ float Bl[2][128 * LDG];
  const int t = threadIdx.x;
  const int lane = t & 31;
  const int w = t >> 5;
  const int wm = w & 3;
  const int wn = w >> 2;
  const int m0 = blockIdx.x * 128;
  const int n0 = blockIdx.y * 128;

  const int seg = t & 3;    // 16B segment within a 64B row-chunk
  const int rbase = t >> 2; // 0..63 (row group; +64 on second pass)

  const int mainChunks = D / 16;
  const int nChunks = mainChunks + 2;  // R == 32 -> two 16-wide tail chunks

  auto stage = [&](int c, int pb) {
#pragma unroll
    for (int g = 0; g < 2; ++g) {
      const int row = rbase + g * 64;
      const float* ga;
      const float* gb;
      if (c < mainChunks) {
        const int k0 = c * 16;
        ga = x_q + (size_t)(m0 + row) * D + k0 + seg * 4;
        gb = wq + (size_t)(n0 + row) * D + k0 + seg * 4;
      } else {
        const int k0 = (c - mainChunks) * 16;
        ga = tbuf + (size_t)(m0 + row) * R + k0 + seg * 4;
        gb = lora_b + (size_t)(n0 + row) * R + k0 + seg * 4;
      }
      async_load_f4(&Al[pb][row * LDG + seg * 4], ga);
      async_load_f4(&Bl[pb][row * LDG + seg * 4], gb);
    }
  };

  v8f acc[2][4] = {};

  stage(0, 0);
  wait_async0();
  __syncthreads();

  const int kb0 = (lane >> 4) << 1;
  const int xl = lane & 15;
  int p = 0;
  for (int c = 0; c < nChunks; ++c) {
    if (c + 1 < nChunks) stage(c + 1, p ^ 1);  // prefetch next chunk (async)

#pragma unroll
    for (int kk = 0; kk < 16; kk += 4) {
      const int kb = kk + kb0;
      v2f a[2], b[4];
#pragma unroll
      for (int i = 0; i < 2; ++i) {
        const int row = wm * 32 + i * 16 + xl;
        a[i].x = Al[p][row * LDG + kb];
        a[i].y = Al[p][row * LDG + kb + 1];
      }
#pragma unroll
      for (int j = 0; j < 4; ++j) {
        const int col = wn * 64 + j * 16 + xl;
        b[j].x = Bl[p][col * LDG + kb];
        b[j].y = Bl[p][col * LDG + kb + 1];
      }
#pragma unroll
      for (int i = 0; i < 2; ++i)
#pragma unroll
        for (int j = 0; j < 4; ++j) acc[i][j] = wmma_f32(a[i], b[j], acc[i][j]);
    }

    wait_async0();   // next chunk landed in LDS
    __syncthreads(); // all waves done reading buf p / async data visible
    p ^= 1;
  }

  // epilogue: + bias, store
  const int mh = (lane >> 4) << 3;
#pragma unroll
  for (int j = 0; j < 4; ++j) {
    const int o = n0 + wn * 64 + j * 16 + xl;
    const float bj = bias[o];
#pragma unroll
    for (int i = 0; i < 2; ++i) {
      const int mbase = m0 + wm * 32 + i * 16 + mh;
#pragma unroll
      for (int e = 0; e < 8; ++e) {
        out[(size_t)(mbase + e) * O + o] = acc[i][j][e] + bj;
      }
    }
  }
}

// ---------------------------------------------------------------------------
extern "C" void kernel_launch(void* const* d_in, const int* in_sizes, int n_in,
                              void* d_out, int out_size, void* d_ws,
                              size_t ws_size, hipStream_t stream) {
  const float* x = (const float*)d_in[0];
  const float* S_in = (const float*)d_in[1];
  // d_in[2] = H_block: unused (FWHT used instead; H is symmetric Sylvester)
  const float* smooth = (const float*)d_in[3];
  const float* wq = (const float*)d_in[4];
  const float* lora_a = (const float*)d_in[5];
  const float* lora_b = (const float*)d_in[6];
  const float* bias = (const float*)d_in[7];

  const int D = in_sizes[1];             // 4096
  const int M = (int)(in_sizes[0] / D);  // B*S = 8192
  const int O = in_sizes[7];             // 4096
  const int R = in_sizes[6] / O;         // 32

  float* x_q = (float*)d_ws;           // M*D
  float* x_sm = x_q + (size_t)M * D;   // M*D
  float* tbuf = x_sm + (size_t)M * D;  // M*R

  k_prep<<<M, 256, D * sizeof(float), stream>>>(x, S_in, smooth, x_q, x_sm, D);
  k_lora<<<M / 128, 256, 0, stream>>>(x_sm, lora_a, tbuf, D);
  k_gemm<<<dim3(M / 128, O / 128), 256, 0, stream>>>(x_q, tbuf, wq, lora_b,
                                                     bias, (float*)d_out, O, D,
                                                     R);
}